// SparseResBlock_25151328485607
// MI455X (gfx1250) — compile-verified
//
#include <hip/hip_runtime.h>

#define CCH 96
#define KOFF 27
#define EPSV 1e-5f
#define ASTRIDE 100   // floats; 400 B rows: 16B-aligned, bank-conflict-free (36*r mod 64 distinct)

typedef __attribute__((ext_vector_type(2))) float v2f;
typedef __attribute__((ext_vector_type(8))) float v8f;
typedef __attribute__((ext_vector_type(4))) int   v4i;

typedef __attribute__((address_space(1))) v4i* gv4ip;   // global (AS1) v4i*
typedef __attribute__((address_space(3))) v4i* lv4ip;   // LDS    (AS3) v4i*

__device__ __forceinline__ void wait_async_zero() {
#if __has_builtin(__builtin_amdgcn_s_wait_asynccnt)
  __builtin_amdgcn_s_wait_asynccnt(0);
#else
  asm volatile("s_wait_asynccnt 0x0" ::: "memory");
#endif
}

// ---------------------------------------------------------------------------
// Zero a float4 buffer
// ---------------------------------------------------------------------------
__global__ __launch_bounds__(256) void zero_f4_kernel(float4* __restrict__ p, int n4) {
  int i = blockIdx.x * blockDim.x + threadIdx.x;
  if (i < n4) p[i] = make_float4(0.f, 0.f, 0.f, 0.f);
}

// ---------------------------------------------------------------------------
// Gather (async global->LDS) -> WMMA(f32 16x16x4) -> atomic scatter.
// Block: 192 threads = 6 wave32; one block = one (k-offset, 16-rule tile).
// Wave w owns output columns [16w, 16w+16).
// ---------------------------------------------------------------------------
__global__ __launch_bounds__(192) void sparse_conv_wmma_kernel(
    const float* __restrict__ feats, const float* __restrict__ W,
    const int* __restrict__ in_idx, const int* __restrict__ out_idx,
    float* __restrict__ acc_out, int M) {
  __shared__ float Ash[16 * ASTRIDE];
  __shared__ int   oidx[16];

  const int t  = threadIdx.x;
  const int k  = blockIdx.y;
  const int m0 = blockIdx.x * 16;
  const int* inI = in_idx + (size_t)k * M + m0;
  const float* Wk = W + (size_t)k * CCH * CCH;

  // Warm WGP$/L2 with the 36 KB weight slice this block will stream 6x24 times.
  #pragma unroll
  for (int pf = 0; pf < 3; ++pf)
    __builtin_prefetch((const char*)Wk + (size_t)t * 64 + (size_t)pf * 12288, 0, 3);

  // Gather 16 scattered feature rows (16 x 96 f32 = 384 float4 chunks) into LDS.
  for (int c = t; c < 16 * 24; c += 192) {
    const int row = c / 24;
    const int col = (c % 24) * 4;
    float* dst = &Ash[row * ASTRIDE + col];
    if (m0 + row < M) {
      const float* src = feats + (size_t)inI[row] * CCH + col;
#if __has_builtin(__builtin_amdgcn_global_load_async_to_lds_b128)
      __builtin_amdgcn_global_load_async_to_lds_b128((gv4ip)src, (lv4ip)dst, 0, 0);
#else
      float4 v = *(const float4*)src;
      dst[0] = v.x; dst[1] = v.y; dst[2] = v.z; dst[3] = v.w;
#endif
    } else {
      dst[0] = 0.f; dst[1] = 0.f; dst[2] = 0.f; dst[3] = 0.f;
    }
  }
  if (t < 16) oidx[t] = (m0 + t < M) ? out_idx[(size_t)k * M + m0 + t] : 0;
#if __has_builtin(__builtin_amdgcn_global_load_async_to_lds_b128)
  wait_async_zero();
#endif
  __syncthreads();

  const int wave = t >> 5;
  const int lane = t & 31;
  const int l16  = lane & 15;
  const int hi   = lane >> 4;        // 0: lanes 0-15, 1: lanes 16-31
  const int n0   = wave * 16;        // output column tile

  // ISA layouts (CDNA5 ISA):
  //   A 16x4: VGPR r, lane L -> M = L%16, K = r + 2*(L>=16)
  //   B 4x16: VGPR r, lane L -> K = r + 2*(L>=16), N = L%16
  const float* arow = &Ash[l16 * ASTRIDE + 2 * hi];
  const float* bcol = Wk + (size_t)(2 * hi) * CCH + n0 + l16;

  v8f acc = {};
  #pragma unroll
  for (int kk = 0; kk < CCH; kk += 4) {
    v2f a, b;
    a.x = arow[kk];
    a.y = arow[kk + 1];
    b.x = bcol[(size_t)kk * CCH];
    b.y = bcol[(size_t)(kk + 1) * CCH];
    acc = __builtin_amdgcn_wmma_f32_16x16x4_f32(
        /*neg_a=*/false, a, /*neg_b=*/false, b,
        /*c_mod=*/(short)0, acc, /*reuse_a=*/false, /*reuse_b=*/false);
  }

  // D 16x16 f32: VGPR r, lane L -> M = r + 8*(L>=16), N = L%16
  #pragma unroll
  for (int r = 0; r < 8; ++r) {
    const int lrow = r + 8 * hi;
    if (m0 + lrow < M) {
      float* dst = acc_out + (size_t)oidx[lrow] * CCH + n0 + l16;
      unsafeAtomicAdd(dst, acc[r]);   // -> global_atomic_add_f32
    }
  }
}

// ---------------------------------------------------------------------------
// y = relu(bn1(y)) in place, and zero the conv2 accumulator buffer.
// ---------------------------------------------------------------------------
__global__ __launch_bounds__(256) void bnrelu_zero_kernel(
    float4* __restrict__ y, float4* __restrict__ zout,
    const float* __restrict__ g, const float* __restrict__ b,
    const float* __restrict__ mu, const float* __restrict__ var, int n4) {
  int i = blockIdx.x * blockDim.x + threadIdx.x;
  if (i >= n4) return;
  const int c0 = (i * 4) % CCH;
  float4 v = y[i];
  float o[4] = {v.x, v.y, v.z, v.w};
  #pragma unroll
  for (int j = 0; j < 4; ++j) {
    const int c = c0 + j;
    const float inv = g[c] * rsqrtf(var[c] + EPSV);
    o[j] = fmaxf(o[j] * inv + (b[c] - mu[c] * inv), 0.f);
  }
  y[i]    = make_float4(o[0], o[1], o[2], o[3]);
  zout[i] = make_float4(0.f, 0.f, 0.f, 0.f);
}

// ---------------------------------------------------------------------------
// out = relu(bn2(out) + feats) in place.
// ---------------------------------------------------------------------------
__global__ __launch_bounds__(256) void bn_res_relu_kernel(
    float4* __restrict__ y, const float4* __restrict__ feats,
    const float* __restrict__ g, const float* __restrict__ b,
    const float* __restrict__ mu, const float* __restrict__ var, int n4) {
  int i = blockIdx.x * blockDim.x + threadIdx.x;
  if (i >= n4) return;
  const int c0 = (i * 4) % CCH;
  float4 v = y[i];
  float4 f = feats[i];
  float o[4] = {v.x, v.y, v.z, v.w};
  float r[4] = {f.x, f.y, f.z, f.w};
  #pragma unroll
  for (int j = 0; j < 4; ++j) {
    const int c = c0 + j;
    const float inv = g[c] * rsqrtf(var[c] + EPSV);
    o[j] = fmaxf(o[j] * inv + (b[c] - mu[c] * inv) + r[j], 0.f);
  }
  y[i] = make_float4(o[0], o[1], o[2], o[3]);
}

// ---------------------------------------------------------------------------
extern "C" void kernel_launch(void* const* d_in, const int* in_sizes, int n_in,
                              void* d_out, int out_size, void* d_ws, size_t ws_size,
                              hipStream_t stream) {
  (void)n_in; (void)out_size; (void)ws_size;

  const float* feats = (const float*)d_in[0];
  const float* w1    = (const float*)d_in[1];
  const float* bn1g  = (const float*)d_in[2];
  const float* bn1b  = (const float*)d_in[3];
  const float* bn1m  = (const float*)d_in[4];
  const float* bn1v  = (const float*)d_in[5];
  const float* w2    = (const float*)d_in[6];
  const float* bn2g  = (const float*)d_in[7];
  const float* bn2b  = (const float*)d_in[8];
  const float* bn2m  = (const float*)d_in[9];
  const float* bn2v  = (const float*)d_in[10];
  const int*   in1   = (const int*)d_in[11];
  const int*   out1  = (const int*)d_in[12];
  const int*   in2   = (const int*)d_in[13];
  const int*   out2  = (const int*)d_in[14];

  const int NC = in_sizes[0];           // N * 96
  const int M  = in_sizes[11] / KOFF;   // rules per offset
  const int n4 = NC / 4;

  float* y1  = (float*)d_ws;            // N x 96 f32 intermediate (76.8 MB)
  float* out = (float*)d_out;

  dim3 egrid((n4 + 255) / 256);
  dim3 cgrid((M + 15) / 16, KOFF);

  zero_f4_kernel<<<egrid, 256, 0, stream>>>((float4*)y1, n4);
  sparse_conv_wmma_kernel<<<cgrid, 192, 0, stream>>>(feats, w1, in1, out1, y1, M);
  bnrelu_zero_kernel<<<egrid, 256, 0, stream>>>((float4*)y1, (float4*)out,
                                                bn1g, bn1b, bn1m, bn1v, n4);
  sparse_conv_wmma_kernel<<<cgrid, 192, 0, stream>>>(y1, w2, in2, out2, out, M);
  bn_res_relu_kernel<<<egrid, 256, 0, stream>>>((float4*)out, (const float4*)feats,
                                                bn2g, bn2b, bn2m, bn2v, n4);
}